// ShootingBlock_53652731461651
// MI455X (gfx1250) — compile-verified
//
#include <hip/hip_runtime.h>

// ---------------------------------------------------------------------------
// Problem constants (from reference): B=4096, K=2048, D=512, T=8, SUBSTEPS=2
// ---------------------------------------------------------------------------
#define D_DIM 512
#define K_DIM 2048
#define B_DIM 4096
#define T_DIM 8

// GEMM tiling
#define MT 128        // M tile per workgroup
#define NT 64         // N tile per workgroup
#define KT 32         // K chunk staged in LDS
#define A_STRIDE 36   // padded LDS stride for A (gcd(36,64)=4 -> 16 lanes conflict-free)

typedef __attribute__((ext_vector_type(2))) float v2f;
typedef __attribute__((ext_vector_type(8))) float v8f;

// ---------------------------------------------------------------------------
// Wave-level 32x32 WMMA micro-kernel helpers (f32, 16x16x4)
// A fragment (16x4, MxK): lane l: m=l&15; VGPR0=K(2*(l>>4)), VGPR1=K(+1)
// B fragment (4x16, KxN): lane l: n=l&15; VGPR0=K(2*(l>>4)), VGPR1=K(+1)
// C/D (16x16): VGPR r: lanes0-15 -> M=r, N=lane; lanes16-31 -> M=r+8, N=lane-16
// ---------------------------------------------------------------------------

__global__ __launch_bounds__(256) void rhs_gemm_kernel(
    const float* __restrict__ Y,      // (B, D) current base state
    const float* __restrict__ Kprev,  // (B, D) previous RK slope
    const float* __restrict__ Wt,     // (D, D) = theta2^T row-major
    const float* __restrict__ bias,   // (D,)
    const float* __restrict__ bt,     // batch_t (T,)
    int iv, float cmul,
    float* __restrict__ Kout)         // (B, D) output slope
{
    __shared__ float As[MT * A_STRIDE];   // 128 x 32 (padded)
    __shared__ float Bs[KT * NT];         // 32 x 64

    const int tid   = threadIdx.x;
    const int lane  = tid & 31;
    const int wave  = tid >> 5;
    const int waveM = wave & 3;           // 4 waves along M (32 rows each)
    const int waveN = wave >> 2;          // 2 waves along N (32 cols each)
    const int b0    = blockIdx.x * MT;
    const int n0    = blockIdx.y * NT;

    const float h = (bt[iv + 1] - bt[iv]) * 0.5f;   // dt / SUBSTEPS
    const float c = h * cmul;                       // stage shift coefficient

    v8f acc[2][2];
    #pragma unroll
    for (int i = 0; i < 2; ++i)
        #pragma unroll
        for (int j = 0; j < 2; ++j)
            #pragma unroll
            for (int r = 0; r < 8; ++r)
                acc[i][j][r] = 0.0f;

    const int mrow = lane & 15;
    const int koff = (lane >> 4) << 1;    // 0 or 2
    const int mbase = waveM * 32;
    const int nbase = waveN * 32;

    for (int k0 = 0; k0 < D_DIM; k0 += KT) {
        // ---- stage A (128x32): relu(Y + c*Kprev), float4 global, scalar LDS ----
        #pragma unroll
        for (int it = 0; it < 4; ++it) {
            int idx = it * 256 + tid;                 // float4 slot 0..1023
            int row = idx >> 3;                       // 0..127
            int cc  = (idx & 7) << 2;                 // 0,4,...,28
            size_t g = (size_t)(b0 + row) * D_DIM + k0 + cc;
            float4 y4 = *(const float4*)(Y + g);
            float4 p4 = *(const float4*)(Kprev + g);
            float* a = As + row * A_STRIDE + cc;
            a[0] = fmaxf(fmaf(c, p4.x, y4.x), 0.0f);
            a[1] = fmaxf(fmaf(c, p4.y, y4.y), 0.0f);
            a[2] = fmaxf(fmaf(c, p4.z, y4.z), 0.0f);
            a[3] = fmaxf(fmaf(c, p4.w, y4.w), 0.0f);
        }
        // ---- stage B (32x64): Wt tile, float4 all the way ----
        #pragma unroll
        for (int it = 0; it < 2; ++it) {
            int idx = it * 256 + tid;                 // float4 slot 0..511
            int r  = idx >> 4;                        // 0..31
            int cc = (idx & 15) << 2;                 // 0..60
            *(float4*)(Bs + r * NT + cc) =
                *(const float4*)(Wt + (size_t)(k0 + r) * D_DIM + n0 + cc);
        }
        __syncthreads();

        #pragma unroll
        for (int kk = 0; kk < KT; kk += 4) {
            v2f a0 = *(const v2f*)(As + (mbase +      mrow) * A_STRIDE + kk + koff);
            v2f a1 = *(const v2f*)(As + (mbase + 16 + mrow) * A_STRIDE + kk + koff);
            v2f b0v, b1v;
            b0v.x = Bs[(kk + koff    ) * NT + nbase +      mrow];
            b0v.y = Bs[(kk + koff + 1) * NT + nbase +      mrow];
            b1v.x = Bs[(kk + koff    ) * NT + nbase + 16 + mrow];
            b1v.y = Bs[(kk + koff + 1) * NT + nbase + 16 + mrow];
            acc[0][0] = __builtin_amdgcn_wmma_f32_16x16x4_f32(false, a0, false, b0v, (short)0, acc[0][0], false, false);
            acc[0][1] = __builtin_amdgcn_wmma_f32_16x16x4_f32(false, a0, false, b1v, (short)0, acc[0][1], false, false);
            acc[1][0] = __builtin_amdgcn_wmma_f32_16x16x4_f32(false, a1, false, b0v, (short)0, acc[1][0], false, false);
            acc[1][1] = __builtin_amdgcn_wmma_f32_16x16x4_f32(false, a1, false, b1v, (short)0, acc[1][1], false, false);
        }
        __syncthreads();
    }

    // ---- epilogue: add bias, store ----
    const int mhi = (lane >> 4) << 3;     // +8 for upper half-wave
    #pragma unroll
    for (int mi = 0; mi < 2; ++mi)
        #pragma unroll
        for (int ni = 0; ni < 2; ++ni) {
            int col = n0 + nbase + ni * 16 + mrow;
            float bv = bias[col];
            #pragma unroll
            for (int r = 0; r < 8; ++r) {
                int row = b0 + mbase + mi * 16 + r + mhi;
                Kout[(size_t)row * D_DIM + col] = acc[mi][ni][r] + bv;
            }
        }
}

// ---------------------------------------------------------------------------
// Phase 1: Wt = theta2^T = -(relu(X)^T @ P), X,P are (K_DIM, D_DIM) row-major.
// M dim = j (columns of X), N dim = i (columns of P), reduction over K_DIM.
// ---------------------------------------------------------------------------
__global__ __launch_bounds__(256) void theta_gemm_kernel(
    const float* __restrict__ X,   // x_params (K, D)
    const float* __restrict__ P,   // p_params (K, D)
    float* __restrict__ Wt)        // (D, D)
{
    __shared__ float As[MT * A_STRIDE];
    __shared__ float Bs[KT * NT];

    const int tid   = threadIdx.x;
    const int lane  = tid & 31;
    const int wave  = tid >> 5;
    const int waveM = wave & 3;
    const int waveN = wave >> 2;
    const int j0    = blockIdx.x * MT;
    const int i0    = blockIdx.y * NT;

    v8f acc[2][2];
    #pragma unroll
    for (int i = 0; i < 2; ++i)
        #pragma unroll
        for (int j = 0; j < 2; ++j)
            #pragma unroll
            for (int r = 0; r < 8; ++r)
                acc[i][j][r] = 0.0f;

    const int mrow = lane & 15;
    const int koff = (lane >> 4) << 1;
    const int mbase = waveM * 32;
    const int nbase = waveN * 32;

    for (int k0 = 0; k0 < K_DIM; k0 += KT) {
        // A[m][kk] = relu(X[k0+kk, j0+m])  (transposed read, coalesced along m)
        #pragma unroll
        for (int it = 0; it < 16; ++it) {
            int idx = it * 256 + tid;      // 0..4095
            int m  = idx & 127;
            int kk = idx >> 7;
            As[m * A_STRIDE + kk] =
                fmaxf(X[(size_t)(k0 + kk) * D_DIM + j0 + m], 0.0f);
        }
        // B[kk][n] = P[k0+kk, i0+n]
        #pragma unroll
        for (int it = 0; it < 2; ++it) {
            int idx = it * 256 + tid;
            int r  = idx >> 4;
            int cc = (idx & 15) << 2;
            *(float4*)(Bs + r * NT + cc) =
                *(const float4*)(P + (size_t)(k0 + r) * D_DIM + i0 + cc);
        }
        __syncthreads();

        #pragma unroll
        for (int kk = 0; kk < KT; kk += 4) {
            v2f a0 = *(const v2f*)(As + (mbase +      mrow) * A_STRIDE + kk + koff);
            v2f a1 = *(const v2f*)(As + (mbase + 16 + mrow) * A_STRIDE + kk + koff);
            v2f b0v, b1v;
            b0v.x = Bs[(kk + koff    ) * NT + nbase +      mrow];
            b0v.y = Bs[(kk + koff + 1) * NT + nbase +      mrow];
            b1v.x = Bs[(kk + koff    ) * NT + nbase + 16 + mrow];
            b1v.y = Bs[(kk + koff + 1) * NT + nbase + 16 + mrow];
            acc[0][0] = __builtin_amdgcn_wmma_f32_16x16x4_f32(false, a0, false, b0v, (short)0, acc[0][0], false, false);
            acc[0][1] = __builtin_amdgcn_wmma_f32_16x16x4_f32(false, a0, false, b1v, (short)0, acc[0][1], false, false);
            acc[1][0] = __builtin_amdgcn_wmma_f32_16x16x4_f32(false, a1, false, b0v, (short)0, acc[1][0], false, false);
            acc[1][1] = __builtin_amdgcn_wmma_f32_16x16x4_f32(false, a1, false, b1v, (short)0, acc[1][1], false, false);
        }
        __syncthreads();
    }

    // Wt[j, i] = -acc  (tile value = +relu(X)^T P)
    const int mhi = (lane >> 4) << 3;
    #pragma unroll
    for (int mi = 0; mi < 2; ++mi)
        #pragma unroll
        for (int ni = 0; ni < 2; ++ni) {
            int col = i0 + nbase + ni * 16 + mrow;
            #pragma unroll
            for (int r = 0; r < 8; ++r) {
                int row = j0 + mbase + mi * 16 + r + mhi;
                Wt[(size_t)row * D_DIM + col] = -acc[mi][ni][r];
            }
        }
}

// bias2[d] = -sum_k P[k,d]
__global__ __launch_bounds__(256) void bias_kernel(
    const float* __restrict__ P, float* __restrict__ bias)
{
    int d = blockIdx.x * blockDim.x + threadIdx.x;
    if (d < D_DIM) {
        float s = 0.0f;
        for (int k = 0; k < K_DIM; ++k) s += P[(size_t)k * D_DIM + d];
        bias[d] = -s;
    }
}

// dst = x + (h * smul) * kv   (float4 vectorized; h derived from batch_t on device)
__global__ __launch_bounds__(256) void axpy_kernel(
    float* __restrict__ dst, const float* __restrict__ x,
    const float* __restrict__ kv, const float* __restrict__ bt,
    int iv, float smul)
{
    const float h = (bt[iv + 1] - bt[iv]) * 0.5f;
    const float s = h * smul;
    size_t i = ((size_t)blockIdx.x * blockDim.x + threadIdx.x) << 2;
    float4 a = *(const float4*)(x + i);
    float4 b = *(const float4*)(kv + i);
    float4 r;
    r.x = fmaf(s, b.x, a.x);
    r.y = fmaf(s, b.y, a.y);
    r.z = fmaf(s, b.z, a.z);
    r.w = fmaf(s, b.w, a.w);
    *(float4*)(dst + i) = r;
}

// ---------------------------------------------------------------------------
extern "C" void kernel_launch(void* const* d_in, const int* in_sizes, int n_in,
                              void* d_out, int out_size, void* d_ws, size_t ws_size,
                              hipStream_t stream)
{
    const float* input   = (const float*)d_in[0];   // (B,1,D)
    const float* bt      = (const float*)d_in[1];   // (T,)
    const float* xparams = (const float*)d_in[2];   // (K,1,D)
    const float* pparams = (const float*)d_in[3];   // (K,1,D)
    // d_in[4], d_in[5] (Mbar, Mbar_b) are dead in the reference (theta/bias unused)

    float* out = (float*)d_out;

    // workspace layout (floats): Wt | bias | Y | ACC | K1 | K2
    float* ws   = (float*)d_ws;
    float* Wt   = ws;
    float* bias = Wt + (size_t)D_DIM * D_DIM;
    float* Y    = bias + D_DIM;                   // offset stays 16B aligned
    const size_t stateN = (size_t)B_DIM * D_DIM;
    float* ACC  = Y + stateN;
    float* K1   = ACC + stateN;
    float* K2   = K1 + stateN;
    const size_t stateBytes = stateN * sizeof(float);

    // Phase 1: parameter contraction
    dim3 gT(D_DIM / MT, D_DIM / NT);              // 4 x 8
    theta_gemm_kernel<<<gT, 256, 0, stream>>>(xparams, pparams, Wt);
    bias_kernel<<<(D_DIM + 255) / 256, 256, 0, stream>>>(pparams, bias);

    // Init state + out[0]
    hipMemcpyAsync(Y,   input, stateBytes, hipMemcpyDeviceToDevice, stream);
    hipMemcpyAsync(out, input, stateBytes, hipMemcpyDeviceToDevice, stream);

    dim3 gR(B_DIM / MT, D_DIM / NT);              // 32 x 8
    const int axpyBlocks = (int)(stateN / 4 / 256);
    const float f16th = 1.0f / 6.0f;
    const float f3rd  = 1.0f / 3.0f;

    for (int iv = 0; iv < T_DIM - 1; ++iv) {
        for (int sub = 0; sub < 2; ++sub) {
            // K1 = rhs(Y)
            rhs_gemm_kernel<<<gR, 256, 0, stream>>>(Y, Y,  Wt, bias, bt, iv, 0.0f, K1);
            // ACC = Y + h/6 * K1
            axpy_kernel<<<axpyBlocks, 256, 0, stream>>>(ACC, Y, K1, bt, iv, f16th);
            // K2 = rhs(Y + h/2 * K1)
            rhs_gemm_kernel<<<gR, 256, 0, stream>>>(Y, K1, Wt, bias, bt, iv, 0.5f, K2);
            // ACC += h/3 * K2
            axpy_kernel<<<axpyBlocks, 256, 0, stream>>>(ACC, ACC, K2, bt, iv, f3rd);
            // K3 = rhs(Y + h/2 * K2)   (reuses K1 buffer)
            rhs_gemm_kernel<<<gR, 256, 0, stream>>>(Y, K2, Wt, bias, bt, iv, 0.5f, K1);
            // ACC += h/3 * K3
            axpy_kernel<<<axpyBlocks, 256, 0, stream>>>(ACC, ACC, K1, bt, iv, f3rd);
            // K4 = rhs(Y + h * K3)     (reuses K2 buffer)
            rhs_gemm_kernel<<<gR, 256, 0, stream>>>(Y, K1, Wt, bias, bt, iv, 1.0f, K2);
            // Y = ACC + h/6 * K4
            axpy_kernel<<<axpyBlocks, 256, 0, stream>>>(Y, ACC, K2, bt, iv, f16th);
        }
        hipMemcpyAsync(out + (size_t)(iv + 1) * stateN, Y, stateBytes,
                       hipMemcpyDeviceToDevice, stream);
    }
}